// LongformerAttention2_77232101917210
// MI455X (gfx1250) — compile-verified
//
#include <hip/hip_runtime.h>
#include <hip/hip_bf16.h>

typedef float v2f __attribute__((ext_vector_type(2)));
typedef float v4f __attribute__((ext_vector_type(4)));
typedef float v8f __attribute__((ext_vector_type(8)));

#define WMMA_F32(a, b, c) \
  __builtin_amdgcn_wmma_f32_16x16x4_f32(false, (a), false, (b), (short)0, (c), false, false)

constexpr int B_  = 2;
constexpr int S_  = 2048;
constexpr int D_  = 1024;
constexpr int H_  = 16;
constexpr int HD_ = 64;
constexpr int W_  = 256;
constexpr int M_  = B_ * S_;  // 4096 rows in the projection GEMMs

// ---------------------------------------------------------------------------
// QKV projection: Y_headmajor[b,h,s,d] = sum_k X[b,s,k] * Wt[h*HD+d][k] + bias
// Block tile 64x64, 8 waves, each wave a 16x32 sub-tile via two f32 WMMA accs.
// ---------------------------------------------------------------------------
__global__ __launch_bounds__(256) void proj_kernel(const float* __restrict__ X,
                                                   const float* __restrict__ Wt,
                                                   const float* __restrict__ bias,
                                                   float* __restrict__ Y) {
  __shared__ float As[64 * 36];  // padded stride 36 (bank-conflict free, 16B aligned)
  __shared__ float Bs[64 * 36];
  const int t = threadIdx.x;
  const int w = t >> 5, lane = t & 31, lh = lane >> 4, ln = lane & 15;
  const int wm = (w & 3) * 16;   // wave row offset in block tile
  const int wn = (w >> 2) * 32;  // wave col offset in block tile
  const int m0 = blockIdx.x * 64, n0 = blockIdx.y * 64;
  v8f acc0 = {}, acc1 = {};
  const int lrow = t >> 2;        // 0..63
  const int lcol = (t & 3) * 8;   // 0,8,16,24

  for (int k0 = 0; k0 < D_; k0 += 32) {
    const float* ap = X + (size_t)(m0 + lrow) * D_ + k0 + lcol;
    const float* bp = Wt + (size_t)(n0 + lrow) * D_ + k0 + lcol;
    *(v4f*)&As[lrow * 36 + lcol]     = *(const v4f*)ap;
    *(v4f*)&As[lrow * 36 + lcol + 4] = *(const v4f*)(ap + 4);
    *(v4f*)&Bs[lrow * 36 + lcol]     = *(const v4f*)bp;
    *(v4f*)&Bs[lrow * 36 + lcol + 4] = *(const v4f*)(bp + 4);
    __syncthreads();
#pragma unroll
    for (int kk = 0; kk < 32; kk += 4) {
      const int kf = kk + 2 * lh;
      v2f a;  a.x  = As[(wm + ln) * 36 + kf];      a.y  = As[(wm + ln) * 36 + kf + 1];
      v2f b0; b0.x = Bs[(wn + ln) * 36 + kf];      b0.y = Bs[(wn + ln) * 36 + kf + 1];
      v2f b1; b1.x = Bs[(wn + 16 + ln) * 36 + kf]; b1.y = Bs[(wn + 16 + ln) * 36 + kf + 1];
      acc0 = WMMA_F32(a, b0, acc0);
      acc1 = WMMA_F32(a, b1, acc1);
    }
    __syncthreads();
  }

#pragma unroll
  for (int r = 0; r < 8; ++r) {
    const int gm = m0 + wm + r + 8 * lh;
    const int bb = gm >> 11, s = gm & (S_ - 1);
    {
      const int gn = n0 + wn + ln;
      const int h = gn >> 6, d = gn & 63;
      Y[(((size_t)bb * H_ + h) * S_ + s) * HD_ + d] = acc0[r] + bias[gn];
    }
    {
      const int gn = n0 + wn + 16 + ln;
      const int h = gn >> 6, d = gn & 63;
      Y[(((size_t)bb * H_ + h) * S_ + s) * HD_ + d] = acc1[r] + bias[gn];
    }
  }
}

// ---------------------------------------------------------------------------
// Output projection: Out[m][n] = sum_k Ctx_headmajor(m,k) * Wo[n][k] + bo[n]
// Same scheme as proj_kernel, A gathered from head-major context.
// ---------------------------------------------------------------------------
__global__ __launch_bounds__(256) void outproj_kernel(const float* Ctx,
                                                      const float* __restrict__ Wo,
                                                      const float* __restrict__ bo,
                                                      float* __restrict__ Out) {
  __shared__ float As[64 * 36];
  __shared__ float Bs[64 * 36];
  const int t = threadIdx.x;
  const int w = t >> 5, lane = t & 31, lh = lane >> 4, ln = lane & 15;
  const int wm = (w & 3) * 16;
  const int wn = (w >> 2) * 32;
  const int m0 = blockIdx.x * 64, n0 = blockIdx.y * 64;
  v8f acc0 = {}, acc1 = {};
  const int lrow = t >> 2;
  const int lcol = (t & 3) * 8;
  const int gmRow = m0 + lrow;
  const int bbr = gmRow >> 11, sr = gmRow & (S_ - 1);

  for (int k0 = 0; k0 < D_; k0 += 32) {
    const int k = k0 + lcol;
    const int h = k >> 6, d = k & 63;  // 8 consecutive k stay within one head
    const float* ap = Ctx + (((size_t)bbr * H_ + h) * S_ + sr) * HD_ + d;
    const float* bp = Wo + (size_t)(n0 + lrow) * D_ + k0 + lcol;
    *(v4f*)&As[lrow * 36 + lcol]     = *(const v4f*)ap;
    *(v4f*)&As[lrow * 36 + lcol + 4] = *(const v4f*)(ap + 4);
    *(v4f*)&Bs[lrow * 36 + lcol]     = *(const v4f*)bp;
    *(v4f*)&Bs[lrow * 36 + lcol + 4] = *(const v4f*)(bp + 4);
    __syncthreads();
#pragma unroll
    for (int kk = 0; kk < 32; kk += 4) {
      const int kf = kk + 2 * lh;
      v2f a;  a.x  = As[(wm + ln) * 36 + kf];      a.y  = As[(wm + ln) * 36 + kf + 1];
      v2f b0; b0.x = Bs[(wn + ln) * 36 + kf];      b0.y = Bs[(wn + ln) * 36 + kf + 1];
      v2f b1; b1.x = Bs[(wn + 16 + ln) * 36 + kf]; b1.y = Bs[(wn + 16 + ln) * 36 + kf + 1];
      acc0 = WMMA_F32(a, b0, acc0);
      acc1 = WMMA_F32(a, b1, acc1);
    }
    __syncthreads();
  }

#pragma unroll
  for (int r = 0; r < 8; ++r) {
    const int gm = m0 + wm + r + 8 * lh;
    {
      const int gn = n0 + wn + ln;
      Out[(size_t)gm * D_ + gn] = acc0[r] + bo[gn];
    }
    {
      const int gn = n0 + wn + 16 + ln;
      Out[(size_t)gm * D_ + gn] = acc1[r] + bo[gn];
    }
  }
}

// ---------------------------------------------------------------------------
// Banded attention: one block per (b, h, 16-query tile).
// scores (WMMA) -> masked LDS -> block softmax -> full attn-row write
// -> context = P @ V (WMMA) -> head-major ctx (aliases q workspace).
// ---------------------------------------------------------------------------
__global__ __launch_bounds__(256) void attn_kernel(const float* Q,
                                                   const float* __restrict__ K,
                                                   const float* __restrict__ V,
                                                   float* __restrict__ Attn,
                                                   float* Ctx) {
  constexpr int NT = 33;          // key tiles covering [q0-256, q0+271]
  constexpr int NK = NT * 16;     // 528 local key slots
  constexpr int SS = 532;         // padded LDS stride for score rows
  constexpr int QS = 68;          // padded LDS stride for Q rows
  __shared__ float lq[16 * QS];
  __shared__ float ls[16 * SS];
  __shared__ float red[256];

  const int t = threadIdx.x, w = t >> 5, lane = t & 31, lh = lane >> 4, ln = lane & 15;
  const int blk = blockIdx.x;
  const int qt = blk & 127;
  const int h  = (blk >> 7) & 15;
  const int bb = blk >> 11;
  const int q0 = qt * 16;
  const int jb = q0 - W_;  // base key index (may be negative)

  const float* qh = Q + (((size_t)bb * H_ + h) * S_) * HD_;
  const float* kh = K + (((size_t)bb * H_ + h) * S_) * HD_;
  const float* vh = V + (((size_t)bb * H_ + h) * S_) * HD_;

  {  // stage Q tile (16 x 64) into LDS
    const int r = t >> 4, c = (t & 15) * 4;
    *(v4f*)&lq[r * QS + c] = *(const v4f*)&qh[(size_t)(q0 + r) * HD_ + c];
  }
  __syncthreads();

  const float scale = 0.125f;  // 1/sqrt(HD)

  // ---- scores: Q (16xHD) @ K^T, per key tile, masked into LDS ----
  for (int lt = w; lt < NT; lt += 8) {
    v8f acc = {};
    const int jrow = jb + lt * 16 + ln;
    const int jc = min(max(jrow, 0), S_ - 1);
    const float* kp = kh + (size_t)jc * HD_;
#pragma unroll
    for (int kk = 0; kk < HD_; kk += 4) {
      const int kf = kk + 2 * lh;
      v2f a; a.x = lq[ln * QS + kf]; a.y = lq[ln * QS + kf + 1];
      v2f b; b.x = kp[kf];           b.y = kp[kf + 1];
      acc = WMMA_F32(a, b, acc);
    }
#pragma unroll
    for (int r = 0; r < 8; ++r) {
      const int m = r + 8 * lh;  // query row within tile
      const int i = q0 + m;
      const int j = jb + lt * 16 + ln;
      const bool ok = (j >= 0) && (j < S_) && (j >= i - W_) && (j <= i + W_);
      ls[m * SS + lt * 16 + ln] = ok ? acc[r] * scale : -__builtin_inff();
    }
  }
  __syncthreads();

  // ---- block softmax over each of the 16 query rows (16 threads/row) ----
  const int r = t >> 4, c0 = t & 15;
  float lmax = -__builtin_inff();
  for (int c = c0; c < NK; c += 16) lmax = fmaxf(lmax, ls[r * SS + c]);
  red[t] = lmax;
  __syncthreads();
  float rmax = -__builtin_inff();
#pragma unroll
  for (int u = 0; u < 16; ++u) rmax = fmaxf(rmax, red[r * 16 + u]);
  float lsum = 0.f;
  for (int c = c0; c < NK; c += 16) {
    const float e = __expf(ls[r * SS + c] - rmax);
    ls[r * SS + c] = e;
    lsum += e;
  }
  __syncthreads();
  red[t] = lsum;
  __syncthreads();
  float rsum = 0.f;
#pragma unroll
  for (int u = 0; u < 16; ++u) rsum += red[r * 16 + u];
  const float inv = 1.0f / rsum;

  // ---- normalize in LDS + write the FULL attn row (zeros outside band) ----
  float* arow = Attn + (((size_t)bb * H_ + h) * S_ + (q0 + r)) * S_;
  for (int jj = c0; jj < S_; jj += 16) {
    const int c = jj - jb;
    float val = 0.f;
    if (c >= 0 && c < NK) {
      const float p = ls[r * SS + c] * inv;
      ls[r * SS + c] = p;
      val = p;
    }
    arow[jj] = val;
  }
  __syncthreads();

  // ---- context = P (16xNK) @ V (NKxHD); waves 0..3, one 16-col tile each ----
  if (w < 4) {
    const int dt = w * 16;
    v8f acc = {};
    for (int kk = 0; kk < NK; kk += 4) {
      const int kf = kk + 2 * lh;
      v2f a; a.x = ls[ln * SS + kf]; a.y = ls[ln * SS + kf + 1];
      const int j0 = jb + kf;
      const int jx = min(max(j0, 0), S_ - 1);
      const int jy = min(max(j0 + 1, 0), S_ - 1);
      v2f b;
      b.x = vh[(size_t)jx * HD_ + dt + ln];
      b.y = vh[(size_t)jy * HD_ + dt + ln];
      acc = WMMA_F32(a, b, acc);
    }
#pragma unroll
    for (int rr = 0; rr < 8; ++rr) {
      const int m = rr + 8 * lh;
      Ctx[(((size_t)bb * H_ + h) * S_ + q0 + m) * HD_ + dt + ln] = acc[rr];
    }
  }
}

// ---------------------------------------------------------------------------
extern "C" void kernel_launch(void* const* d_in, const int* in_sizes, int n_in,
                              void* d_out, int out_size, void* d_ws, size_t ws_size,
                              hipStream_t stream) {
  const float* x  = (const float*)d_in[0];
  const float* Wq = (const float*)d_in[1];
  const float* bq = (const float*)d_in[2];
  const float* Wk = (const float*)d_in[3];
  const float* bk = (const float*)d_in[4];
  const float* Wv = (const float*)d_in[5];
  const float* bv = (const float*)d_in[6];
  const float* Wo = (const float*)d_in[7];
  const float* bo = (const float*)d_in[8];

  float* out  = (float*)d_out;
  float* attn = out + (size_t)B_ * S_ * D_;

  // workspace: q, k, v head-major [B,H,S,HD]; context aliases q (safe:
  // each attn block consumes its own 16 q-rows via LDS before overwriting).
  float* q = (float*)d_ws;
  float* k = q + (size_t)B_ * S_ * D_;
  float* v = k + (size_t)B_ * S_ * D_;
  float* ctx = q;

  dim3 g(M_ / 64, D_ / 64);
  proj_kernel<<<g, 256, 0, stream>>>(x, Wq, bq, q);
  proj_kernel<<<g, 256, 0, stream>>>(x, Wk, bk, k);
  proj_kernel<<<g, 256, 0, stream>>>(x, Wv, bv, v);

  attn_kernel<<<B_ * H_ * (S_ / 16), 256, 0, stream>>>(q, k, v, attn, ctx);

  outproj_kernel<<<g, 256, 0, stream>>>(ctx, Wo, bo, out);
}